// EncoderLayer_72378788872457
// MI455X (gfx1250) — compile-verified
//
#include <hip/hip_runtime.h>
#include <math.h>
#include <stdint.h>

typedef __attribute__((ext_vector_type(2))) float v2f;
typedef __attribute__((ext_vector_type(8))) float v8f;

#define WMMA_F32(a, b, c) \
  __builtin_amdgcn_wmma_f32_16x16x4_f32(false, (a), false, (b), (short)0, (c), false, false)

static constexpr int Bb  = 4;
static constexpr int Nn  = 2048;
static constexpr int Dd  = 512;
static constexpr int Hh  = 8;
static constexpr int Kk  = 64;
static constexpr int DFF = 2048;
static constexpr int BN  = Bb * Nn;          // 8192
static constexpr int QKVS = 3 * Hh * Kk;     // 1536: row stride of fused qkv

// ---------------------------------------------------------------------------
// CDNA5 async global->LDS copy (ASYNCcnt-tracked, no VGPR round trip).
// LDS address = low 32 bits of the generic pointer.
// ---------------------------------------------------------------------------
__device__ __forceinline__ void async_ld_b128(const void* gptr, void* lptr) {
  uint32_t lds = (uint32_t)(uintptr_t)lptr;
  asm volatile("global_load_async_to_lds_b128 %0, %1, off"
               :: "v"(lds), "v"(gptr) : "memory");
}
__device__ __forceinline__ void wait_async_all() {
  asm volatile("s_wait_asynccnt 0x0" ::: "memory");
}
__device__ __forceinline__ void wait_async_16() {
  asm volatile("s_wait_asynccnt 0x10" ::: "memory");
}

// ---------------------------------------------------------------------------
// Tiled WMMA GEMM: C[M x N] = A[M x Kd] * B[Kd x N] (+bias, +relu)
// 128 threads = 4 waves; block tile 64x64; each wave owns a 32x32 sub-tile.
// Double-buffered async LDS staging overlaps copy of tile t+1 with compute of t.
// ---------------------------------------------------------------------------
__global__ __launch_bounds__(128) void gemm_wmma(
    const float* __restrict__ A, const float* __restrict__ Bm,
    float* __restrict__ C, int M, int N, int Kd,
    const float* __restrict__ bias, int relu) {
  // strides: A=20 (80B rows, 16B aligned, conflict-free b64 frags)
  //          B=72 (288B rows, 16B aligned, lane-halves on disjoint banks)
  __shared__ __align__(16) float As[2][64][20];
  __shared__ __align__(16) float Bs[2][16][72];

  const int tid  = threadIdx.x;
  const int lane = tid & 31;
  const int w    = tid >> 5;
  const int row0 = blockIdx.y * 64;
  const int col0 = blockIdx.x * 64;
  const int rw   = (w >> 1) * 32;
  const int cw   = (w & 1) * 32;
  const int c16  = lane & 15;
  const int k0   = (lane >> 4) * 2;

  // Stage one 64x16 A tile + 16x64 B tile: 4 async b128 per thread, no loops.
  auto stage = [&](int buf, int kt) {
    {
      int r = tid >> 2, c4 = (tid & 3) * 4;            // rows 0..31
      async_ld_b128(A + (size_t)(row0 + r) * Kd + kt + c4, &As[buf][r][c4]);
      r += 32;                                          // rows 32..63
      async_ld_b128(A + (size_t)(row0 + r) * Kd + kt + c4, &As[buf][r][c4]);
    }
    {
      int kk = tid >> 4, c4 = (tid & 15) * 4;          // k rows 0..7
      async_ld_b128(Bm + (size_t)(kt + kk) * N + col0 + c4, &Bs[buf][kk][c4]);
      kk += 8;                                          // k rows 8..15
      async_ld_b128(Bm + (size_t)(kt + kk) * N + col0 + c4, &Bs[buf][kk][c4]);
    }
  };

  v8f acc[2][2] = {};

  stage(0, 0);
  wait_async_all();
  __syncthreads();

  int buf = 0;
  for (int kt = 0; kt < Kd; kt += 16) {
    const bool more = (kt + 16) < Kd;
    if (more) stage(buf ^ 1, kt + 16);                  // overlaps with compute

#pragma unroll
    for (int s = 0; s < 4; ++s) {
      const int kk = s * 4 + k0;
      v2f a0 = *(const v2f*)&As[buf][rw + c16][kk];
      v2f a1 = *(const v2f*)&As[buf][rw + 16 + c16][kk];
      v2f b0 = { Bs[buf][kk][cw + c16],      Bs[buf][kk + 1][cw + c16] };
      v2f b1 = { Bs[buf][kk][cw + 16 + c16], Bs[buf][kk + 1][cw + 16 + c16] };
      acc[0][0] = WMMA_F32(a0, b0, acc[0][0]);
      acc[0][1] = WMMA_F32(a0, b1, acc[0][1]);
      acc[1][0] = WMMA_F32(a1, b0, acc[1][0]);
      acc[1][1] = WMMA_F32(a1, b1, acc[1][1]);
    }

    if (more) {
      wait_async_all();          // only the tile-(t+1) copies remain outstanding
      __syncthreads();
    }
    buf ^= 1;
  }

  // Epilogue: C layout = lane holds (row i + 8*(lane>>4), col lane&15) per VGPR i
#pragma unroll
  for (int ti = 0; ti < 2; ++ti) {
#pragma unroll
    for (int tj = 0; tj < 2; ++tj) {
      const int colg = col0 + cw + tj * 16 + c16;
      const float bv = bias ? bias[colg] : 0.0f;
#pragma unroll
      for (int i = 0; i < 8; ++i) {
        const int rowg = row0 + rw + ti * 16 + i + 8 * (lane >> 4);
        float v = acc[ti][tj][i] + bv;
        if (relu) v = fmaxf(v, 0.0f);
        C[(size_t)rowg * N + colg] = v;
      }
    }
  }
}

// ---------------------------------------------------------------------------
// Weight repacking
// ---------------------------------------------------------------------------
__global__ void pack_qkv_w(const float* __restrict__ Pq, const float* __restrict__ Pk,
                           const float* __restrict__ Pv, float* __restrict__ W) {
  int idx = blockIdx.x * 256 + threadIdx.x;             // W is (D, 3*H*K)
  if (idx >= Dd * QKVS) return;
  int d   = idx / QKVS;
  int rem = idx % QKVS;
  int p   = rem / (Hh * Kk);
  int hk  = rem % (Hh * Kk);
  int h   = hk >> 6, k = hk & 63;
  const float* P = (p == 0) ? Pq : ((p == 1) ? Pk : Pv);
  W[idx] = P[((size_t)h * Dd + d) * Kk + k];
}

__global__ void pack_wo(const float* __restrict__ Po, float* __restrict__ Wo) {
  int idx = blockIdx.x * 256 + threadIdx.x;             // Wo is (H*K, D)
  if (idx >= Hh * Kk * Dd) return;
  int row = idx >> 9;                                   // h*64 + v
  int d   = idx & 511;
  int h   = row >> 6, v = row & 63;
  Wo[idx] = Po[((size_t)h * Dd + d) * Kk + v];
}

// ---------------------------------------------------------------------------
// RoPE, in-place on q and k sections of fused qkv buffer
// ---------------------------------------------------------------------------
__global__ __launch_bounds__(256) void rope_kernel(float* __restrict__ qkv,
                                                   const int* __restrict__ positions) {
  const int bn = blockIdx.x;
  const int t  = threadIdx.x;
  const int h  = t >> 5, j = t & 31;
  const float pos  = (float)positions[bn];
  const float freq = (float)j * (2.0f / (float)Kk);
  const float ts   = powf(10000.0f, freq);
  const float ang  = pos / ts;
  float sn, cs;
  sincosf(ang, &sn, &cs);

  float* qrow = qkv + (size_t)bn * QKVS + h * Kk;
  float x1 = qrow[j], x2 = qrow[j + 32];
  qrow[j]      = x1 * cs - x2 * sn;
  qrow[j + 32] = x2 * cs + x1 * sn;

  float* krow = qrow + Hh * Kk;
  x1 = krow[j]; x2 = krow[j + 32];
  krow[j]      = x1 * cs - x2 * sn;
  krow[j + 32] = x2 * cs + x1 * sn;
}

// ---------------------------------------------------------------------------
// Flash attention: 1 wave per 16-row Q block per (b,h). Double-buffered async
// K/V tile staging; since the workgroup is a single wave, no barriers needed —
// just ASYNCcnt waits (in-order completion => s_wait_asynccnt 16 frees tile t
// while tile t+1 is still in flight).
// ---------------------------------------------------------------------------
__global__ __launch_bounds__(32) void attn_flash(const float* __restrict__ qkv,
                                                 float* __restrict__ ctx) {
  const int bid  = blockIdx.x;
  const int nt   = bid & (Nn / 16 - 1);
  const int h    = (bid >> 7) & (Hh - 1);
  const int b    = bid >> 10;
  const int lane = threadIdx.x;
  const int c16  = lane & 15;
  const int k0   = (lane >> 4) * 2;

  __shared__ __align__(16) float Kl[2][16][68];   // [key][kdim] row-major
  __shared__ __align__(16) float Vl[2][16][72];   // [key][vcol] row-major
  __shared__ __align__(16) float Pl[16][18];      // [qrow][key]

  const int n0 = nt * 16;
  const float* qbase = qkv + (size_t)(b * Nn) * QKVS + h * Kk;
  const float* kbase = qbase + Hh * Kk;
  const float* vbase = qbase + 2 * Hh * Kk;

  // 16 async b128 per K/V tile pair (2 rows per instruction; lane-half = row parity)
  auto stage_kv = [&](int buf, int j0) {
    const int c4 = (lane & 15) * 4;
    const int rp = lane >> 4;
#pragma unroll
    for (int i = 0; i < 8; ++i) {
      const int r = 2 * i + rp;
      async_ld_b128(kbase + (size_t)(j0 + r) * QKVS + c4, &Kl[buf][r][c4]);
      async_ld_b128(vbase + (size_t)(j0 + r) * QKVS + c4, &Vl[buf][r][c4]);
    }
  };

  // Preload Q fragments: 16 k-steps of 4
  v2f qf[16];
  {
    const float* qrow = qbase + (size_t)(n0 + c16) * QKVS;
#pragma unroll
    for (int s = 0; s < 16; ++s)
      qf[s] = *(const v2f*)(qrow + 4 * s + k0);
  }

  float mrow[8], lrow[8];
  v8f o[4] = {};
#pragma unroll
  for (int i = 0; i < 8; ++i) { mrow[i] = -3.0e38f; lrow[i] = 0.0f; }

  stage_kv(0, 0);

  for (int jt = 0; jt < Nn / 16; ++jt) {
    const int buf = jt & 1;
    const bool more = (jt + 1) < (Nn / 16);
    if (more) stage_kv(buf ^ 1, (jt + 1) * 16);   // prefetch next tile
    if (more) wait_async_16();                    // current tile's 16 ops are done
    else      wait_async_all();

    // S = Q * K^T  (B[kk][c] = K[c][kk] -> contiguous b64 reads of Kl rows)
    v8f sacc = {};
#pragma unroll
    for (int s = 0; s < 16; ++s) {
      v2f bf = *(const v2f*)&Kl[buf][c16][4 * s + k0];
      sacc = WMMA_F32(qf[s], bf, sacc);
    }

    // Online softmax (rows: i -> row i + 8*(lane>>4); xor masks stay in 16-lane halves)
#pragma unroll
    for (int i = 0; i < 8; ++i) {
      float v = sacc[i];
      float rmax = v;
      rmax = fmaxf(rmax, __shfl_xor(rmax, 1, 32));
      rmax = fmaxf(rmax, __shfl_xor(rmax, 2, 32));
      rmax = fmaxf(rmax, __shfl_xor(rmax, 4, 32));
      rmax = fmaxf(rmax, __shfl_xor(rmax, 8, 32));
      const float mnew  = fmaxf(mrow[i], rmax);
      const float alpha = __expf(mrow[i] - mnew);
      const float p     = __expf(v - mnew);
      float psum = p;
      psum += __shfl_xor(psum, 1, 32);
      psum += __shfl_xor(psum, 2, 32);
      psum += __shfl_xor(psum, 4, 32);
      psum += __shfl_xor(psum, 8, 32);
      lrow[i] = lrow[i] * alpha + psum;
      mrow[i] = mnew;
#pragma unroll
      for (int vb = 0; vb < 4; ++vb) o[vb][i] *= alpha;
      Pl[i + 8 * (lane >> 4)][c16] = p;           // C-layout -> LDS -> A-layout
    }

    // O += P * V  (4 k-steps x 4 column tiles); V row-major: paired b32 frags
#pragma unroll
    for (int s = 0; s < 4; ++s) {
      const int kk = 4 * s + k0;
      v2f a = *(const v2f*)&Pl[c16][kk];
#pragma unroll
      for (int vb = 0; vb < 4; ++vb) {
        v2f bf = { Vl[buf][kk][vb * 16 + c16], Vl[buf][kk + 1][vb * 16 + c16] };
        o[vb] = WMMA_F32(a, bf, o[vb]);
      }
    }
  }

  // Normalize and write ctx in (B*N, H*K) row-major layout
#pragma unroll
  for (int i = 0; i < 8; ++i) {
    const float inv = 1.0f / lrow[i];
    const int m = i + 8 * (lane >> 4);
    float* orow = ctx + (size_t)(b * Nn + n0 + m) * (Hh * Kk) + h * Kk;
#pragma unroll
    for (int vb = 0; vb < 4; ++vb)
      orow[vb * 16 + c16] = o[vb][i] * inv;
  }
}

// ---------------------------------------------------------------------------
// out = LayerNorm(a*amul + b*bmul) with scale/bias. One block (256 thr) per row.
// ---------------------------------------------------------------------------
__global__ __launch_bounds__(256) void add_ln_kernel(
    const float* __restrict__ a, const float* __restrict__ bsrc,
    float amul, float bmul,
    const float* __restrict__ scale, const float* __restrict__ bias,
    float* __restrict__ out) {
  const int row = blockIdx.x;
  const int t   = threadIdx.x;
  const float* ar = a + (size_t)row * Dd;
  float v0 = ar[t] * amul;
  float v1 = ar[t + 256] * amul;
  if (bsrc) {
    const float* br = bsrc + (size_t)row * Dd;
    v0 += br[t] * bmul;
    v1 += br[t + 256] * bmul;
  }
  __shared__ float red[8];
  float s = v0 + v1;
#pragma unroll
  for (int off = 16; off; off >>= 1) s += __shfl_xor(s, off, 32);
  if ((t & 31) == 0) red[t >> 5] = s;
  __syncthreads();
  float tot = 0.0f;
#pragma unroll
  for (int i = 0; i < 8; ++i) tot += red[i];
  const float mu = tot * (1.0f / (float)Dd);
  const float d0 = v0 - mu, d1 = v1 - mu;
  __syncthreads();
  float sq = d0 * d0 + d1 * d1;
#pragma unroll
  for (int off = 16; off; off >>= 1) sq += __shfl_xor(sq, off, 32);
  if ((t & 31) == 0) red[t >> 5] = sq;
  __syncthreads();
  float var = 0.0f;
#pragma unroll
  for (int i = 0; i < 8; ++i) var += red[i];
  var *= (1.0f / (float)Dd);
  const float rs = rsqrtf(var + 1e-6f);
  out[(size_t)row * Dd + t]       = d0 * rs * scale[t] + bias[t];
  out[(size_t)row * Dd + t + 256] = d1 * rs * scale[t + 256] + bias[t + 256];
}

// ---------------------------------------------------------------------------
extern "C" void kernel_launch(void* const* d_in, const int* in_sizes, int n_in,
                              void* d_out, int out_size, void* d_ws, size_t ws_size,
                              hipStream_t stream) {
  const float* x    = (const float*)d_in[0];
  const int*   pos  = (const int*)d_in[1];
  const float* Pq   = (const float*)d_in[2];
  const float* Pk   = (const float*)d_in[3];
  const float* Pv   = (const float*)d_in[4];
  const float* Po   = (const float*)d_in[5];
  const float* W1   = (const float*)d_in[6];
  const float* b1   = (const float*)d_in[7];
  const float* W2   = (const float*)d_in[8];
  const float* b2   = (const float*)d_in[9];
  const float* ln1s = (const float*)d_in[10];
  const float* ln1b = (const float*)d_in[11];
  const float* ln2s = (const float*)d_in[12];
  const float* ln2b = (const float*)d_in[13];

  float* ws   = (float*)d_ws;
  float* qkv  = ws;                    // 8192*1536 = 12,582,912
  float* Wqkv = ws + 12582912;         //  786,432
  float* Wo   = ws + 13369344;         //  262,144
  float* ctx  = ws + 13631488;         // 4,194,304
  float* attn = ws + 17825792;         // 4,194,304
  float* h1   = ws + 22020096;         // 4,194,304  (total 100 MB)
  float* mid  = ws;                    // FFN mid reuses qkv..ctx (dead by then)
  float* ffn  = attn;                  // FFN out reuses attn (dead after LN1)

  pack_qkv_w<<<(Dd * QKVS + 255) / 256, 256, 0, stream>>>(Pq, Pk, Pv, Wqkv);
  pack_wo<<<(Hh * Kk * Dd + 255) / 256, 256, 0, stream>>>(Po, Wo);

  gemm_wmma<<<dim3(QKVS / 64, BN / 64), 128, 0, stream>>>(
      x, Wqkv, qkv, BN, QKVS, Dd, nullptr, 0);

  rope_kernel<<<BN, 256, 0, stream>>>(qkv, pos);

  attn_flash<<<Bb * Hh * (Nn / 16), 32, 0, stream>>>(qkv, ctx);

  gemm_wmma<<<dim3(Dd / 64, BN / 64), 128, 0, stream>>>(
      ctx, Wo, attn, BN, Dd, Dd, nullptr, 0);

  add_ln_kernel<<<BN, 256, 0, stream>>>(x, attn, 1.0f, 1.0f, ln1s, ln1b, h1);

  gemm_wmma<<<dim3(DFF / 64, BN / 64), 128, 0, stream>>>(
      h1, W1, mid, BN, DFF, Dd, b1, 1);

  gemm_wmma<<<dim3(Dd / 64, BN / 64), 128, 0, stream>>>(
      mid, W2, ffn, BN, Dd, DFF, b2, 0);

  add_ln_kernel<<<BN, 256, 0, stream>>>(ffn, nullptr, 2.0f, 0.0f, ln2s, ln2b,
                                        (float*)d_out);
}